// GSOccLocalBridge_5849745457879
// MI455X (gfx1250) — compile-verified
//
#include <hip/hip_runtime.h>
#include <hip/hip_bf16.h>

typedef __attribute__((ext_vector_type(2))) float v2f;
typedef __attribute__((ext_vector_type(4))) float v4f;
typedef __attribute__((ext_vector_type(8))) float v8f;

#define BTC      2
#define NPTS_PER 16896      // V*H*W = 6*32*88
#define NPTS     33792      // BT * NPTS_PER
#define CG       48
#define CO       256
#define NZV      10
#define NYV      100
#define NXV      100
#define NVOXC    100000
#define OUT1_ELEMS ((size_t)BTC * CO * NVOXC)   // 51,200,000 (multiple of 4)
#define CNT_ELEMS  ((size_t)BTC * NVOXC)        // 200,000    (multiple of 4)

static __device__ __forceinline__ v8f wmma_f32_4(v2f a, v2f b, v8f c) {
    // D = A(16x4 f32) * B(4x16 f32) + C(16x16 f32)  -> v_wmma_f32_16x16x4_f32
    return __builtin_amdgcn_wmma_f32_16x16x4_f32(false, a, false, b, (short)0, c,
                                                 false, false);
}

// ------------------------------------------------- zero init (b128 stores)
__global__ void k_zero(v4f* __restrict__ out1, v4f* __restrict__ cnt) {
    size_t i = (size_t)blockIdx.x * blockDim.x + threadIdx.x;
    v4f z = {};
    if (i < OUT1_ELEMS / 4) out1[i] = z;
    if (i < CNT_ELEMS / 4) cnt[i] = z;
}

// ------------------------------------------------------ voxel index + counts
__global__ void k_voxelize(const float* __restrict__ centers,
                           int* __restrict__ linidx, float* __restrict__ cnt) {
    int p = blockIdx.x * blockDim.x + threadIdx.x;
    if (p >= NPTS) return;
    float x = centers[p * 3 + 0];
    float y = centers[p * 3 + 1];
    float z = centers[p * 3 + 2];
    int ix = (int)floorf((x + 40.0f) * 1.25f);   // /0.8
    int iy = (int)floorf((y + 40.0f) * 1.25f);
    int iz = (int)floorf((z + 2.0f) * 1.25f);
    bool valid = (ix >= 0) && (ix < NXV) && (iy >= 0) && (iy < NYV) &&
                 (iz >= 0) && (iz < NZV);
    int xc = min(max(ix, 0), NXV - 1);
    int yc = min(max(iy, 0), NYV - 1);
    int zc = min(max(iz, 0), NZV - 1);
    int lin = zc * (NYV * NXV) + yc * NXV + xc;
    int bt = p / NPTS_PER;
    if (valid) {
        linidx[p] = lin;
        atomicAdd(cnt + (size_t)bt * NVOXC + lin, 1.0f);
    } else {
        linidx[p] = -1;
    }
}

// ------------------------------- GEMM feats@W_g2o + b, scatter-add to voxels
// one wave => one 16(M)x16(N) tile; 2112 M-tiles x 16 N-tiles = 33792 waves
__global__ void k_g2o(const float* __restrict__ feats,
                      const float* __restrict__ Wg, const float* __restrict__ bg,
                      const int* __restrict__ linidx, float* __restrict__ out1) {
    __shared__ float lW[CG * CO];   // 48KB
    for (int i = threadIdx.x; i < CG * CO; i += 256) lW[i] = Wg[i];
    __syncthreads();

    int l    = threadIdx.x & 31;
    int wid  = threadIdx.x >> 5;
    int gw   = blockIdx.x * 8 + wid;
    int mtile = gw >> 4;
    int ntile = gw & 15;
    int half = l >> 4;
    int lm   = l & 15;
    int n    = ntile * 16 + lm;

    const float* arow = feats + (size_t)(mtile * 16 + lm) * CG;
    v8f c = {};
#pragma unroll
    for (int kk = 0; kk < 12; ++kk) {
        int k = kk * 4 + half * 2;
        v2f a, b;
        a.x = arow[k];
        a.y = arow[k + 1];
        b.x = lW[k * CO + n];
        b.y = lW[(k + 1) * CO + n];
        c = wmma_f32_4(a, b, c);
    }
    float bias = bg[n];
#pragma unroll
    for (int r = 0; r < 8; ++r) {
        int row = half * 8 + r;
        int p   = mtile * 16 + row;
        int bt  = p / NPTS_PER;
        int lin = linidx[p];
        if (lin >= 0)
            atomicAdd(out1 + ((size_t)(bt * CO + n)) * NVOXC + lin, c[r] + bias);
    }
}

// ---------------------------- divide sums by counts (b128 load/store pairs)
__global__ void k_normalize(v4f* __restrict__ out1, const v4f* __restrict__ cnt) {
    size_t i = (size_t)blockIdx.x * blockDim.x + threadIdx.x;
    if (i >= OUT1_ELEMS / 4) return;
    size_t e   = i * 4;
    size_t v   = e % NVOXC;           // multiple of 4; never crosses a row
    size_t btc = e / NVOXC;
    size_t bt  = btc / CO;
    v4f cc = cnt[(bt * NVOXC + v) / 4];
    v4f o  = out1[i];
    o.x = o.x / fmaxf(cc.x, 1.0f);
    o.y = o.y / fmaxf(cc.y, 1.0f);
    o.z = o.z / fmaxf(cc.z, 1.0f);
    o.w = o.w / fmaxf(cc.w, 1.0f);
    out1[i] = o;
}

// --------------- trilinear sample (K=256) fused with GEMM sampled@W_o2g + b
// one wave => 16 points x 48 outputs (3 N-tiles); 2112 waves total
__global__ void k_o2g(const float* __restrict__ centers,
                      const float* __restrict__ occ,
                      const float* __restrict__ Wo, const float* __restrict__ bo,
                      float* __restrict__ out2) {
    __shared__ float lW[CO * CG];   // 48KB
    for (int i = threadIdx.x; i < CO * CG; i += 256) lW[i] = Wo[i];
    __syncthreads();

    int l     = threadIdx.x & 31;
    int wid   = threadIdx.x >> 5;
    int mtile = blockIdx.x * 8 + wid;
    int half  = l >> 4;
    int lm    = l & 15;
    int p     = mtile * 16 + lm;   // this lane's point (duplicated per half)
    int bt    = p / NPTS_PER;

    float px = centers[p * 3 + 0];
    float py = centers[p * 3 + 1];
    float pz = centers[p * 3 + 2];
    // pix = (pt - grid_min)/VOXEL - 0.5
    float fx = (px + 40.0f) * 1.25f - 0.5f;
    float fy = (py + 40.0f) * 1.25f - 0.5f;
    float fz = (pz + 2.0f) * 1.25f - 0.5f;
    float f0x = floorf(fx), f0y = floorf(fy), f0z = floorf(fz);
    float frx = fx - f0x, fry = fy - f0y, frz = fz - f0z;
    int x0 = (int)f0x, y0 = (int)f0y, z0 = (int)f0z;

    float w[8];
    int   li[8];
#pragma unroll
    for (int j = 0; j < 8; ++j) {
        int dx = j & 1, dy = (j >> 1) & 1, dz = (j >> 2) & 1;
        int xi = x0 + dx, yi = y0 + dy, zi = z0 + dz;
        bool inb = (xi >= 0) && (xi < NXV) && (yi >= 0) && (yi < NYV) &&
                   (zi >= 0) && (zi < NZV);
        float ww = (dx ? frx : 1.0f - frx) * (dy ? fry : 1.0f - fry) *
                   (dz ? frz : 1.0f - frz);
        w[j] = inb ? ww : 0.0f;
        int xc = min(max(xi, 0), NXV - 1);
        int yc = min(max(yi, 0), NYV - 1);
        int zc = min(max(zi, 0), NZV - 1);
        li[j] = zc * (NYV * NXV) + yc * NXV + xc;
    }

    const float* obase = occ + (size_t)bt * CO * NVOXC;
    v8f c0 = {}, c1 = {}, c2 = {};
    for (int kk = 0; kk < 64; ++kk) {
        int k = kk * 4 + half * 2;
        const float* ch0 = obase + (size_t)k * NVOXC;
        const float* ch1 = ch0 + NVOXC;
        float s0 = 0.0f, s1 = 0.0f;
#pragma unroll
        for (int j = 0; j < 8; ++j) {
            s0 += w[j] * ch0[li[j]];
            s1 += w[j] * ch1[li[j]];
        }
        v2f a;
        a.x = s0;
        a.y = s1;
        v2f b0, b1, b2;
        b0.x = lW[k * CG + lm];           b0.y = lW[(k + 1) * CG + lm];
        b1.x = lW[k * CG + 16 + lm];      b1.y = lW[(k + 1) * CG + 16 + lm];
        b2.x = lW[k * CG + 32 + lm];      b2.y = lW[(k + 1) * CG + 32 + lm];
        c0 = wmma_f32_4(a, b0, c0);
        c1 = wmma_f32_4(a, b1, c1);
        c2 = wmma_f32_4(a, b2, c2);
    }

#pragma unroll
    for (int nt = 0; nt < 3; ++nt) {
        v8f cc = (nt == 0) ? c0 : ((nt == 1) ? c1 : c2);
        int n = nt * 16 + lm;
        float bias = bo[n];
#pragma unroll
        for (int r = 0; r < 8; ++r) {
            int row = half * 8 + r;
            int pp  = mtile * 16 + row;
            out2[(size_t)pp * CG + n] = cc[r] + bias;
        }
    }
}

extern "C" void kernel_launch(void* const* d_in, const int* in_sizes, int n_in,
                              void* d_out, int out_size, void* d_ws, size_t ws_size,
                              hipStream_t stream) {
    const float* centers = (const float*)d_in[0];
    const float* feats   = (const float*)d_in[1];
    const float* occ     = (const float*)d_in[2];
    const float* Wg2o    = (const float*)d_in[3];
    const float* bg2o    = (const float*)d_in[4];
    const float* Wo2g    = (const float*)d_in[5];
    const float* bo2g    = (const float*)d_in[6];

    float* out1 = (float*)d_out;                 // [BT, Co, NVOX]
    float* out2 = out1 + OUT1_ELEMS;             // [BT*N, Cg]

    float* cnt    = (float*)d_ws;                                        // 800 KB
    int*   linidx = (int*)((char*)d_ws + CNT_ELEMS * sizeof(float));

    // 1) zero accumulators (vectorized b128)
    {
        size_t n4 = OUT1_ELEMS / 4;
        int blocks = (int)((n4 + 255) / 256);
        k_zero<<<blocks, 256, 0, stream>>>((v4f*)out1, (v4f*)cnt);
    }
    // 2) voxelize points + counts
    k_voxelize<<<(NPTS + 255) / 256, 256, 0, stream>>>(centers, linidx, cnt);
    // 3) WMMA GEMM + scatter-add  (33792 waves / 8 per block)
    k_g2o<<<(NPTS / 16) * (CO / 16) / 8, 256, 0, stream>>>(feats, Wg2o, bg2o,
                                                           linidx, out1);
    // 4) normalize to mean (vectorized b128)
    {
        size_t n4 = OUT1_ELEMS / 4;
        int blocks = (int)((n4 + 255) / 256);
        k_normalize<<<blocks, 256, 0, stream>>>((v4f*)out1, (const v4f*)cnt);
    }
    // 5) trilinear sample + WMMA GEMM to out2  (2112 waves / 8 per block)
    k_o2g<<<(NPTS / 16) / 8, 256, 0, stream>>>(centers, occ, Wo2g, bo2g, out2);
}